// EquivariantLayerNorm_43147241456119
// MI455X (gfx1250) — compile-verified
//
#include <hip/hip_runtime.h>

typedef float v2f __attribute__((ext_vector_type(2)));
typedef float v4f __attribute__((ext_vector_type(4)));
typedef float v8f __attribute__((ext_vector_type(8)));

#define WAVES_PER_BLOCK 8
#define ROWS_PER_WAVE 5   // 5 rows * 3 vectors = 15 M-rows, M=15 is the all-ones row
#define D_DIM 256
#define EPS_D 1.0e-3

__device__ __forceinline__ double newton_polish(double l, double tr, double m2, double det) {
#pragma unroll
  for (int it = 0; it < 2; ++it) {
    double pv = ((l - tr) * l + m2) * l - det;   // char poly  l^3 - tr l^2 + m2 l - det
    double dv = (3.0 * l - 2.0 * tr) * l + m2;
    double step = (fabs(dv) > 1e-300) ? (pv / dv) : 0.0;
    l -= step;
  }
  return l;
}

__global__ __launch_bounds__(256) void eqln_wmma_kernel(const float* __restrict__ input,
                                                        const float* __restrict__ weight,
                                                        float* __restrict__ outp, int N) {
  __shared__ float ldsG[WAVES_PER_BLOCK][256];                 // 16x16 Gram per wave
  __shared__ float ldsM[WAVES_PER_BLOCK][ROWS_PER_WAVE][10];   // Minv(6) + mean(3)

  const int w    = threadIdx.x >> 5;
  const int lane = threadIdx.x & 31;
  const long group = (long)blockIdx.x * WAVES_PER_BLOCK + w;
  const long base  = group * ROWS_PER_WAVE;

  // ---------------- Phase A: Gram via V_WMMA_F32_16X16X4_F32 ----------------
  // A (16x4 f32) layout: lanes 0-15 = M rows, VGPR0 holds K=k (lanes<16) / K=k+2
  // (lanes>=16); VGPR1 holds K=k+1 / K=k+3.  The 4x16 B layout mirrors this, so
  // feeding the same registers as A and B computes D += A * A^T (Gram).
  const int m     = lane & 15;            // M row handled by this lane
  const int khalf = (lane >> 4) << 1;     // 0 or 2
  long rowA = base + (m / 3);
  if (rowA >= N) rowA = N - 1;            // clamp: safe dummy loads for tail/pad
  const int vec = m - (m / 3) * 3;
  const float* pA = input + ((rowA * 3 + vec) * (long)D_DIM + khalf);
  const bool isOnes = (m == 15);          // ones row -> Gram row 15 = column sums

  v8f acc = {};
#pragma unroll 8
  for (int k = 0; k < D_DIM; k += 4) {
    v2f xv = *(const v2f*)(pA + k);
    xv.x = isOnes ? 1.0f : xv.x;
    xv.y = isOnes ? 1.0f : xv.y;
    // D = A*B + C ; args: (neg_a, A, neg_b, B, c_mod, C, reuse_a, reuse_b)
    acc = __builtin_amdgcn_wmma_f32_16x16x4_f32(false, xv, false, xv, (short)0, acc,
                                                false, false);
  }

  // C/D layout: VGPR i, lanes 0-15 -> (M=i, N=lane); lanes 16-31 -> (M=i+8, N=lane-16)
  {
    const int rowOff = (lane < 16) ? 0 : 8;
    const int colOff = lane & 15;
#pragma unroll
    for (int i = 0; i < 8; ++i) ldsG[w][(i + rowOff) * 16 + colOff] = acc[i];
  }
  __syncthreads();

  // ---------------- Phase B: per-row 3x3 inverse-sqrt (fp64), lanes 0..4 ----
  {
    const int a = lane;
    if (a < ROWS_PER_WAVE && (base + a) < N) {
      const float* G = ldsG[w];
      const int o = 3 * a;
      const double invD = 1.0 / (double)D_DIM;
      const double g00 = G[(o + 0) * 16 + (o + 0)];
      const double g01 = G[(o + 0) * 16 + (o + 1)];
      const double g02 = G[(o + 0) * 16 + (o + 2)];
      const double g11 = G[(o + 1) * 16 + (o + 1)];
      const double g12 = G[(o + 1) * 16 + (o + 2)];
      const double g22 = G[(o + 2) * 16 + (o + 2)];
      const double s0 = G[15 * 16 + (o + 0)];
      const double s1 = G[15 * 16 + (o + 1)];
      const double s2 = G[15 * 16 + (o + 2)];
      const double mu0 = s0 * invD, mu1 = s1 * invD, mu2 = s2 * invD;
      // centered covariance + eps * diag(1,2,3)
      const double c00 = (g00 - s0 * s0 * invD) * invD + EPS_D * 1.0;
      const double c01 = (g01 - s0 * s1 * invD) * invD;
      const double c02 = (g02 - s0 * s2 * invD) * invD;
      const double c11 = (g11 - s1 * s1 * invD) * invD + EPS_D * 2.0;
      const double c12 = (g12 - s1 * s2 * invD) * invD;
      const double c22 = (g22 - s2 * s2 * invD) * invD + EPS_D * 3.0;

      // eigenvalues of symmetric 3x3 (trig closed form, f32 seed + fp64 polish)
      const double q  = (c00 + c11 + c22) * (1.0 / 3.0);
      const double p1 = c01 * c01 + c02 * c02 + c12 * c12;
      const double b00 = c00 - q, b11 = c11 - q, b22 = c22 - q;
      const double p2 = b00 * b00 + b11 * b11 + b22 * b22 + 2.0 * p1;
      double lam0, lam1, lam2;
      if (p2 < 1e-24) {
        lam0 = lam1 = lam2 = q;
      } else {
        const double p  = sqrt(p2 * (1.0 / 6.0));
        const double ip = 1.0 / p;
        const double e00 = b00 * ip, e11 = b11 * ip, e22 = b22 * ip;
        const double e01 = c01 * ip, e02 = c02 * ip, e12 = c12 * ip;
        double r = 0.5 * (e00 * (e11 * e22 - e12 * e12)
                        - e01 * (e01 * e22 - e12 * e02)
                        + e02 * (e01 * e12 - e11 * e02));
        r = fmin(1.0, fmax(-1.0, r));
        const float phi = acosf((float)r) * (1.0f / 3.0f);
        double l0 = q + 2.0 * p * (double)cosf(phi);
        double l2 = q + 2.0 * p * (double)cosf(phi + 2.09439510239319549f);
        double l1 = 3.0 * q - l0 - l2;
        const double tr  = c00 + c11 + c22;
        const double m2c = c00 * c11 - c01 * c01 + c00 * c22 - c02 * c02
                         + c11 * c22 - c12 * c12;
        const double det = c00 * (c11 * c22 - c12 * c12)
                         - c01 * (c01 * c22 - c12 * c02)
                         + c02 * (c01 * c12 - c11 * c02);
        lam0 = newton_polish(l0, tr, m2c, det);
        lam1 = newton_polish(l1, tr, m2c, det);
        lam2 = newton_polish(l2, tr, m2c, det);
      }

      // f(C) = (C + eps I)^{-1/2} applied per eigenvalue: f(l)=1/sqrt(l+eps).
      // Newton divided-difference form; all terms algebraically cancellation-free.
      const double sA = sqrt(fmax(lam0, 0.0) + EPS_D);
      const double sB = sqrt(fmax(lam1, 0.0) + EPS_D);
      const double sC = sqrt(fmax(lam2, 0.0) + EPS_D);
      const double d0 = 1.0 / sA;
      const double d1 = -1.0 / (sA * sB * (sA + sB));
      const double d2 = (sA + sB + sC) /
                        (sA * sB * sC * (sA + sB) * (sB + sC) * (sA + sC));
      // C^2 (symmetric)
      const double cc00 = c00 * c00 + c01 * c01 + c02 * c02;
      const double cc01 = c00 * c01 + c01 * c11 + c02 * c12;
      const double cc02 = c00 * c02 + c01 * c12 + c02 * c22;
      const double cc11 = c01 * c01 + c11 * c11 + c12 * c12;
      const double cc12 = c01 * c02 + c11 * c12 + c12 * c22;
      const double cc22 = c02 * c02 + c12 * c12 + c22 * c22;
      const double t1 = lam0 + lam1, t0 = lam0 * lam1;
      float* Mo = ldsM[w][a];
      Mo[0] = (float)(d0 + d1 * (c00 - lam0) + d2 * (cc00 - t1 * c00 + t0));
      Mo[1] = (float)(     d1 * c01          + d2 * (cc01 - t1 * c01));
      Mo[2] = (float)(     d1 * c02          + d2 * (cc02 - t1 * c02));
      Mo[3] = (float)(d0 + d1 * (c11 - lam0) + d2 * (cc11 - t1 * c11 + t0));
      Mo[4] = (float)(     d1 * c12          + d2 * (cc12 - t1 * c12));
      Mo[5] = (float)(d0 + d1 * (c22 - lam0) + d2 * (cc22 - t1 * c22 + t0));
      Mo[6] = (float)mu0;
      Mo[7] = (float)mu1;
      Mo[8] = (float)mu2;
    }
  }
  __syncthreads();

  // ---------------- Phase C: apply M to centered x (input re-read hits L2) ---
#pragma unroll
  for (int a2 = 0; a2 < ROWS_PER_WAVE; ++a2) {
    const long row = base + a2;
    if (row < N) {
      const float* Mp = ldsM[w][a2];
      const float M00 = Mp[0], M01 = Mp[1], M02 = Mp[2];
      const float M11 = Mp[3], M12 = Mp[4], M22 = Mp[5];
      const float mu0 = Mp[6], mu1 = Mp[7], mu2 = Mp[8];
      const int j = lane * 8;
      const float* px = input + row * 3 * (long)D_DIM;
      v4f xa0 = *(const v4f*)(px + 0 * D_DIM + j);
      v4f xb0 = *(const v4f*)(px + 0 * D_DIM + j + 4);
      v4f xa1 = *(const v4f*)(px + 1 * D_DIM + j);
      v4f xb1 = *(const v4f*)(px + 1 * D_DIM + j + 4);
      v4f xa2 = *(const v4f*)(px + 2 * D_DIM + j);
      v4f xb2 = *(const v4f*)(px + 2 * D_DIM + j + 4);
      v4f wa = *(const v4f*)(weight + j);
      v4f wb = *(const v4f*)(weight + j + 4);
      v4f ya0, ya1, ya2, yb0, yb1, yb2;
#pragma unroll
      for (int e = 0; e < 4; ++e) {
        const float c0 = xa0[e] - mu0;
        const float c1 = xa1[e] - mu1;
        const float c2 = xa2[e] - mu2;
        const float ww = wa[e];
        ya0[e] = (M00 * c0 + M01 * c1 + M02 * c2) * ww;
        ya1[e] = (M01 * c0 + M11 * c1 + M12 * c2) * ww;
        ya2[e] = (M02 * c0 + M12 * c1 + M22 * c2) * ww;
      }
#pragma unroll
      for (int e = 0; e < 4; ++e) {
        const float c0 = xb0[e] - mu0;
        const float c1 = xb1[e] - mu1;
        const float c2 = xb2[e] - mu2;
        const float ww = wb[e];
        yb0[e] = (M00 * c0 + M01 * c1 + M02 * c2) * ww;
        yb1[e] = (M01 * c0 + M11 * c1 + M12 * c2) * ww;
        yb2[e] = (M02 * c0 + M12 * c1 + M22 * c2) * ww;
      }
      float* po = outp + row * 3 * (long)D_DIM;
      *(v4f*)(po + 0 * D_DIM + j)     = ya0;
      *(v4f*)(po + 0 * D_DIM + j + 4) = yb0;
      *(v4f*)(po + 1 * D_DIM + j)     = ya1;
      *(v4f*)(po + 1 * D_DIM + j + 4) = yb1;
      *(v4f*)(po + 2 * D_DIM + j)     = ya2;
      *(v4f*)(po + 2 * D_DIM + j + 4) = yb2;
    }
  }
}

extern "C" void kernel_launch(void* const* d_in, const int* in_sizes, int n_in,
                              void* d_out, int out_size, void* d_ws, size_t ws_size,
                              hipStream_t stream) {
  const float* input  = (const float*)d_in[0];
  const float* weight = (const float*)d_in[1];
  float* outp = (float*)d_out;
  const int N = in_sizes[0] / (3 * D_DIM);
  const long groups = ((long)N + ROWS_PER_WAVE - 1) / ROWS_PER_WAVE;
  const int blocks = (int)((groups + WAVES_PER_BLOCK - 1) / WAVES_PER_BLOCK);
  eqln_wmma_kernel<<<blocks, 256, 0, stream>>>(input, weight, outp, N);
}